// Model_13984413516166
// MI455X (gfx1250) — compile-verified
//
#include <hip/hip_runtime.h>
#include <math.h>

typedef __attribute__((ext_vector_type(2))) float v2f;
typedef __attribute__((ext_vector_type(8))) float v8f;

#define C_DIM 128
#define H_DIM 512
#define CH_DIM 256
#define NPG_   128
#define CHUNK_F (512 * 33)   // LDS floats for one 512x32 tile, stride 33 (bank-conflict-free)

// ---------------- degree / normalization ----------------
__global__ void k_init_deg(float* deg, int n) {
  int i = blockIdx.x * blockDim.x + threadIdx.x;
  if (i < n) deg[i] = 1.0f;  // self-loop weight
}

__global__ void k_deg_accum(const int* __restrict__ ei, const float* __restrict__ w,
                            float* deg, int E) {
  int e = blockIdx.x * blockDim.x + threadIdx.x;
  if (e < E) atomicAdd(&deg[ei[E + e]], w[e]);
}

__global__ void k_dinv(float* deg, int n) {
  int i = blockIdx.x * blockDim.x + threadIdx.x;
  if (i < n) {
    float d = deg[i];
    deg[i] = d > 0.0f ? rsqrtf(d) : 0.0f;
  }
}

// ------------- pack A (row-major MxK) into wmma-f32 fragment order, 2 k-steps/float4
// float4 i = ((mt*(K/8)+ks2)*32 + lane):
//   .xy = A[m][ks2*8 + 2*(lane>>4) + {0,1}]      (k-step 2*ks2)
//   .zw = A[m][ks2*8 + 4 + 2*(lane>>4) + {0,1}]  (k-step 2*ks2+1)
__global__ void k_pack_a4(const float* __restrict__ A, float* __restrict__ out,
                          int M, int K, int total) {
  int i = blockIdx.x * blockDim.x + threadIdx.x;
  if (i >= total) return;
  int lane = i & 31;
  int rest = i >> 5;
  int k8 = K >> 3;
  int ks2 = rest % k8;
  int mt = rest / k8;
  int m = (mt << 4) + (lane & 15);
  int k0 = (ks2 << 3) + ((lane >> 4) << 1);
  const float* row = A + (size_t)m * K;
  float4 v;
  v.x = row[k0];
  v.y = row[k0 + 1];
  v.z = row[k0 + 4];
  v.w = row[k0 + 5];
  ((float4*)out)[i] = v;
}

// ------------- pack B (row-major KxN) into wmma-f32 fragment order, 2 k-steps/float4
// float4 i = ((ntg*(K/8)+ks2)*32 + lane):
//   .xy = B[ks2*8 + 2*(lane>>4) + {0,1}][ntg*16 + (lane&15)]
//   .zw = B[ks2*8 + 4 + 2*(lane>>4) + {0,1}][ntg*16 + (lane&15)]
__global__ void k_pack_b4(const float* __restrict__ B, float* __restrict__ out,
                          int N, int K, int total) {
  int i = blockIdx.x * blockDim.x + threadIdx.x;
  if (i >= total) return;
  int lane = i & 31;
  int rest = i >> 5;
  int k8 = K >> 3;
  int ks2 = rest % k8;
  int ntg = rest / k8;
  int n = (ntg << 4) + (lane & 15);
  int k0 = (ks2 << 3) + ((lane >> 4) << 1);
  float4 v;
  v.x = B[(size_t)k0 * N + n];
  v.y = B[(size_t)(k0 + 1) * N + n];
  v.z = B[(size_t)(k0 + 4) * N + n];
  v.w = B[(size_t)(k0 + 5) * N + n];
  ((float4*)out)[i] = v;
}

// ---------------- GEMM: C(MxN row-major) = packedA * packedB --------------------
// one wave computes a 16x64 tile; software-pipelined b128 fragment loads
__global__ void k_gemm_wmma(const float* __restrict__ pA, const float* __restrict__ pB,
                            float* __restrict__ Cc, int N, int K, int tiles) {
  int wave = threadIdx.x >> 5;
  int lane = threadIdx.x & 31;
  int tile = blockIdx.x * (blockDim.x >> 5) + wave;
  if (tile >= tiles) return;
  int nblk = N >> 6;
  int mt = tile / nblk;
  int nb = tile % nblk;
  int k8 = K >> 3;
  v8f acc[4] = {};
  const float4* ap = (const float4*)pA + (size_t)mt * k8 * 32 + lane;
  const float4* bp = (const float4*)pB + (size_t)(nb << 2) * k8 * 32 + lane;

  float4 a_c = ap[0];
  float4 b_c[4];
#pragma unroll
  for (int nt = 0; nt < 4; ++nt) b_c[nt] = bp[(size_t)nt * k8 * 32];

  for (int ks2 = 0; ks2 < k8; ++ks2) {
    int nxt = (ks2 + 1 < k8) ? (ks2 + 1) : ks2;  // clamped prefetch (no branch)
    float4 a_n = ap[(size_t)nxt * 32];
    float4 b_n[4];
#pragma unroll
    for (int nt = 0; nt < 4; ++nt) b_n[nt] = bp[((size_t)nt * k8 + nxt) * 32];

    v2f a0; a0.x = a_c.x; a0.y = a_c.y;
    v2f a1; a1.x = a_c.z; a1.y = a_c.w;
#pragma unroll
    for (int nt = 0; nt < 4; ++nt) {
      v2f b0; b0.x = b_c[nt].x; b0.y = b_c[nt].y;
      acc[nt] = __builtin_amdgcn_wmma_f32_16x16x4_f32(false, a0, false, b0, (short)0,
                                                      acc[nt], false, false);
    }
#pragma unroll
    for (int nt = 0; nt < 4; ++nt) {
      v2f b1; b1.x = b_c[nt].z; b1.y = b_c[nt].w;
      acc[nt] = __builtin_amdgcn_wmma_f32_16x16x4_f32(false, a1, false, b1, (short)0,
                                                      acc[nt], false, false);
    }
    a_c = a_n;
#pragma unroll
    for (int nt = 0; nt < 4; ++nt) b_c[nt] = b_n[nt];
  }

  int nlo = lane & 15;
  int rowhi = (lane >> 4) << 3;
#pragma unroll
  for (int nt = 0; nt < 4; ++nt)
#pragma unroll
    for (int r = 0; r < 8; ++r) {
      int row = (mt << 4) + r + rowhi;
      Cc[(size_t)row * N + (nb << 6) + nt * 16 + nlo] = acc[nt][r];
    }
}

// ---------------- self-loop init: acc[n][:] = T[n][:] * dinv[n]^2 ----------------
__global__ void k_selfloop(const float* __restrict__ T, const float* __restrict__ dinv,
                           float* __restrict__ acc, int total4) {
  int i = blockIdx.x * blockDim.x + threadIdx.x;
  if (i >= total4) return;
  int node = i >> 7;  // H/4 = 128 float4 per node
  float di = dinv[node];
  float s = di * di;
  float4 t = ((const float4*)T)[i];
  ((float4*)acc)[i] = make_float4(t.x * s, t.y * s, t.z * s, t.w * s);
}

// ---------------- SpMM edge scatter: one wave per edge, 512 features -------------
__global__ void k_spmm(const int* __restrict__ ei, const float* __restrict__ w,
                       const float* __restrict__ dinv, const float* __restrict__ T,
                       float* __restrict__ acc, int E) {
  int wave = threadIdx.x >> 5;
  int lane = threadIdx.x & 31;
  int e = blockIdx.x * (blockDim.x >> 5) + wave;
  if (e >= E) return;
  int s = ei[e];
  int d = ei[E + e];
  float coef = dinv[s] * w[e] * dinv[d];
  const float4* ts = (const float4*)(T + (size_t)s * H_DIM);
  float* ad = acc + (size_t)d * H_DIM;
#pragma unroll
  for (int j = 0; j < 4; ++j) {
    int idx = j * 32 + lane;  // coalesced float4 per wave
    float4 v = ts[idx];
    float* p = ad + idx * 4;
    atomicAdd(p + 0, v.x * coef);
    atomicAdd(p + 1, v.y * coef);
    atomicAdd(p + 2, v.z * coef);
    atomicAdd(p + 3, v.w * coef);
  }
}

// ---------------- fused head: tanh+bias -> cnn1 -> relu -> cnn2 -> relu -> lin ---
// one workgroup (256 threads = 8 waves) per graph; activations staged in LDS
__global__ void __launch_bounds__(256) k_head(
    const float* __restrict__ accS, const float* __restrict__ b_gcn,
    const float* __restrict__ pW1, const float* __restrict__ b1,
    const float* __restrict__ pW2, const float* __restrict__ b2,
    const float* __restrict__ w_lin, const float* __restrict__ b_lin,
    float* __restrict__ out) {
  extern __shared__ float lds[];
  float* chunk = lds;             // 512*33 floats
  float* red = lds + CHUNK_F;     // 256 floats
  int b = blockIdx.x;
  int tid = threadIdx.x;
  int wave = tid >> 5;
  int lane = tid & 31;
  int nlo = lane & 15;
  int halfsel = (lane >> 4) << 1;
  int rowhi = (lane >> 4) << 3;
  const float4* pW1v = (const float4*)pW1;
  const float4* pW2v = (const float4*)pW2;

  // ---- cnn1: Y1(512x32) = relu(W1(512x2048) @ Hb(2048x32) + b1), K streamed in 4 chunks
  v8f acc1[4][2] = {};
  for (int g = 0; g < 4; ++g) {
    __syncthreads();  // protect previous chunk readers
    const float* src = accS + ((size_t)(b * NPG_ + g * 32)) * H_DIM;
    for (int idx = tid; idx < 512 * 32; idx += 256) {
      int f = idx & 511;      // coalesced along f
      int ne = idx >> 9;      // node-in-group = e position
      chunk[f * 33 + ne] = tanhf(src[(size_t)ne * H_DIM + f] + b_gcn[f]);
    }
    __syncthreads();
    for (int ks2 = 0; ks2 < 64; ++ks2) {   // 2 k-steps per iteration
      int ks2g = (g << 6) + ks2;           // global ks2 over K=2048 (K/8 = 256)
      float4 a4[4];
#pragma unroll
      for (int t = 0; t < 4; ++t) {
        int ot = (wave << 2) + t;  // o-tile 0..31
        a4[t] = pW1v[((size_t)(ot * 256 + ks2g)) * 32 + lane];
      }
      int kk0 = (ks2 << 3) + halfsel;
      v2f b0[2], b1f[2];
#pragma unroll
      for (int et = 0; et < 2; ++et) {
        int n = (et << 4) + nlo;
        b0[et].x = chunk[kk0 * 33 + n];
        b0[et].y = chunk[(kk0 + 1) * 33 + n];
        b1f[et].x = chunk[(kk0 + 4) * 33 + n];
        b1f[et].y = chunk[(kk0 + 5) * 33 + n];
      }
#pragma unroll
      for (int t = 0; t < 4; ++t) {
        v2f a0; a0.x = a4[t].x; a0.y = a4[t].y;
        v2f a1; a1.x = a4[t].z; a1.y = a4[t].w;
#pragma unroll
        for (int et = 0; et < 2; ++et) {
          acc1[t][et] = __builtin_amdgcn_wmma_f32_16x16x4_f32(
              false, a0, false, b0[et], (short)0, acc1[t][et], false, false);
          acc1[t][et] = __builtin_amdgcn_wmma_f32_16x16x4_f32(
              false, a1, false, b1f[et], (short)0, acc1[t][et], false, false);
        }
      }
    }
  }
  __syncthreads();
  // bias + relu, Y1 -> LDS (reuse chunk buffer, same padded layout)
#pragma unroll
  for (int t = 0; t < 4; ++t)
#pragma unroll
    for (int et = 0; et < 2; ++et)
#pragma unroll
      for (int r = 0; r < 8; ++r) {
        int o = (wave << 6) + (t << 4) + r + rowhi;
        int e = (et << 4) + nlo;
        float v = acc1[t][et][r] + b1[o];
        chunk[o * 33 + e] = v > 0.0f ? v : 0.0f;
      }
  __syncthreads();

  // ---- cnn2: Y2(256x32) = relu(W2(256x512) @ Y1 + b2), fused with w_lin dot
  v8f acc2[2][2] = {};
  for (int ks2 = 0; ks2 < 64; ++ks2) {  // K=512 -> K/8 = 64
    float4 a4[2];
#pragma unroll
    for (int t = 0; t < 2; ++t) {
      int ot = (wave << 1) + t;  // o-tile 0..15
      a4[t] = pW2v[((size_t)(ot * 64 + ks2)) * 32 + lane];
    }
    int kk0 = (ks2 << 3) + halfsel;
    v2f b0[2], b1f[2];
#pragma unroll
    for (int et = 0; et < 2; ++et) {
      int n = (et << 4) + nlo;
      b0[et].x = chunk[kk0 * 33 + n];
      b0[et].y = chunk[(kk0 + 1) * 33 + n];
      b1f[et].x = chunk[(kk0 + 4) * 33 + n];
      b1f[et].y = chunk[(kk0 + 5) * 33 + n];
    }
#pragma unroll
    for (int t = 0; t < 2; ++t) {
      v2f a0; a0.x = a4[t].x; a0.y = a4[t].y;
      v2f a1; a1.x = a4[t].z; a1.y = a4[t].w;
#pragma unroll
      for (int et = 0; et < 2; ++et) {
        acc2[t][et] = __builtin_amdgcn_wmma_f32_16x16x4_f32(
            false, a0, false, b0[et], (short)0, acc2[t][et], false, false);
        acc2[t][et] = __builtin_amdgcn_wmma_f32_16x16x4_f32(
            false, a1, false, b1f[et], (short)0, acc2[t][et], false, false);
      }
    }
  }
  float partial = 0.0f;
#pragma unroll
  for (int t = 0; t < 2; ++t)
#pragma unroll
    for (int et = 0; et < 2; ++et)
#pragma unroll
      for (int r = 0; r < 8; ++r) {
        int o = (wave << 5) + (t << 4) + r + rowhi;
        int e = (et << 4) + nlo;
        float v = acc2[t][et][r] + b2[o];
        v = v > 0.0f ? v : 0.0f;
        partial += v * w_lin[o * 32 + e];
      }
  red[tid] = partial;
  __syncthreads();
  for (int s = 128; s > 0; s >>= 1) {
    if (tid < s) red[tid] += red[tid + s];
    __syncthreads();
  }
  if (tid == 0) {
    float z = red[0] + b_lin[0];
    out[b] = 1.0f / (1.0f + expf(-z));
  }
}

// =================================================================================
extern "C" void kernel_launch(void* const* d_in, const int* in_sizes, int n_in,
                              void* d_out, int out_size, void* d_ws, size_t ws_size,
                              hipStream_t stream) {
  const float* x         = (const float*)d_in[0];
  const float* edge_attr = (const float*)d_in[1];
  const float* w1_gcn2   = (const float*)d_in[2];
  const float* w_gcn     = (const float*)d_in[3];
  const float* b_gcn     = (const float*)d_in[4];
  const float* w_cnn1    = (const float*)d_in[5];
  const float* b_cnn1    = (const float*)d_in[6];
  const float* w_cnn2    = (const float*)d_in[7];
  const float* b_cnn2    = (const float*)d_in[8];
  const float* w_lin     = (const float*)d_in[9];
  const float* b_lin     = (const float*)d_in[10];
  const int*   ei        = (const int*)d_in[11];
  float* out = (float*)d_out;

  int N = in_sizes[0] / C_DIM;  // 65536
  int E = in_sizes[1];          // 524288
  int BS = N / NPG_;            // 512

  char* ws = (char*)d_ws;
  size_t off = 0;
  auto wsalloc = [&](size_t bytes) -> void* {
    void* p = ws + off;
    off += (bytes + 255) & ~(size_t)255;
    return p;
  };
  float* dinv = (float*)wsalloc((size_t)N * 4);
  float* pw1a = (float*)wsalloc((size_t)C_DIM * C_DIM * 4);
  float* pwgb = (float*)wsalloc((size_t)C_DIM * H_DIM * 4);
  float* Wc   = (float*)wsalloc((size_t)C_DIM * H_DIM * 4);
  float* pWcb = (float*)wsalloc((size_t)C_DIM * H_DIM * 4);
  float* pxa  = (float*)wsalloc((size_t)N * C_DIM * 4);
  float* T    = (float*)wsalloc((size_t)N * H_DIM * 4);
  float* accb = (float*)wsalloc((size_t)N * H_DIM * 4);
  float* pc1  = (float*)wsalloc((size_t)H_DIM * 4 * H_DIM * 4);
  float* pc2  = (float*)wsalloc((size_t)CH_DIM * H_DIM * 4);
  (void)ws_size;

  // 1) symmetric normalization (with self-loops)
  k_init_deg<<<(N + 255) / 256, 256, 0, stream>>>(dinv, N);
  k_deg_accum<<<(E + 255) / 256, 256, 0, stream>>>(ei, edge_attr, dinv, E);
  k_dinv<<<(N + 255) / 256, 256, 0, stream>>>(dinv, N);

  // 2) Wc = w1_gcn2 @ w_gcn (GCN2Conv with alpha=beta=1 collapses to x @ W1)
  {
    int ta = (C_DIM / 16) * (C_DIM / 8) * 32;
    k_pack_a4<<<(ta + 255) / 256, 256, 0, stream>>>(w1_gcn2, pw1a, C_DIM, C_DIM, ta);
    int tb = (H_DIM / 16) * (C_DIM / 8) * 32;
    k_pack_b4<<<(tb + 255) / 256, 256, 0, stream>>>(w_gcn, pwgb, H_DIM, C_DIM, tb);
    int tiles = (C_DIM / 16) * (H_DIM / 64);
    k_gemm_wmma<<<(tiles + 3) / 4, 128, 0, stream>>>(pw1a, pwgb, Wc, H_DIM, C_DIM, tiles);
  }
  // 3) T = x @ Wc  (65536 x 128 x 512, fp32 WMMA)
  {
    int tb = (H_DIM / 16) * (C_DIM / 8) * 32;
    k_pack_b4<<<(tb + 255) / 256, 256, 0, stream>>>(Wc, pWcb, H_DIM, C_DIM, tb);
    int ta = (N / 16) * (C_DIM / 8) * 32;
    k_pack_a4<<<(ta + 255) / 256, 256, 0, stream>>>(x, pxa, N, C_DIM, ta);
    int tiles = (N / 16) * (H_DIM / 64);
    k_gemm_wmma<<<(tiles + 3) / 4, 128, 0, stream>>>(pxa, pWcb, T, H_DIM, C_DIM, tiles);
  }
  // 4) acc = A_hat @ T (self loop + edge scatter)
  {
    int total4 = N * (H_DIM / 4);
    k_selfloop<<<(total4 + 255) / 256, 256, 0, stream>>>(T, dinv, accb, total4);
    k_spmm<<<(E + 7) / 8, 256, 0, stream>>>(ei, edge_attr, dinv, T, accb, E);
  }
  // 5) pack conv weights into fragment order
  {
    int ta = (H_DIM / 16) * ((4 * H_DIM) / 8) * 32;
    k_pack_a4<<<(ta + 255) / 256, 256, 0, stream>>>(w_cnn1, pc1, H_DIM, 4 * H_DIM, ta);
  }
  {
    int ta = (CH_DIM / 16) * (H_DIM / 8) * 32;
    k_pack_a4<<<(ta + 255) / 256, 256, 0, stream>>>(w_cnn2, pc2, CH_DIM, H_DIM, ta);
  }
  // 6) fused head per graph
  k_head<<<BS, 256, (CHUNK_F + 256) * 4, stream>>>(accb, b_gcn, pc1, b_cnn1, pc2,
                                                   b_cnn2, w_lin, b_lin, out);
}